// ISNELayer_68822555951155
// MI455X (gfx1250) — compile-verified
//
#include <hip/hip_runtime.h>
#include <stdint.h>

// Problem constants (match reference)
#define N_NODES 100000
#define HIDDEN  128            // floats per row
#define H4      (HIDDEN / 4)   // float4 per row = 32 -> one wave32 covers a row
#define WAVES_PER_BLOCK 8      // 256 threads

// ---------------------------------------------------------------------------
// CSR path (atomic-free float reduction)
// ws layout (ints):
//   [0,            N)            counts
//   [N,            2N+1)         offsets (N+1)
//   [2N+1,         3N+1)         cursors
//   [3N+1,         3N+1+E)       sorted_src
// ---------------------------------------------------------------------------

__global__ void zero_counts_kernel(int* __restrict__ counts, int n) {
    int i = blockIdx.x * blockDim.x + threadIdx.x;
    if (i < n) counts[i] = 0;
}

__global__ void hist_kernel(const int* __restrict__ edge_index, int* __restrict__ counts, int E) {
    int i = blockIdx.x * blockDim.x + threadIdx.x;
    if (i < E) {
        int tgt = edge_index[E + i];   // row 1 of [2,E]
        atomicAdd(&counts[tgt], 1);
    }
}

// Single-workgroup exclusive scan of counts -> offsets (N+1) and cursors (N).
__global__ void scan_kernel(const int* __restrict__ counts,
                            int* __restrict__ offsets,
                            int* __restrict__ cursors, int n) {
    __shared__ int partial[1024];
    const int t = threadIdx.x;
    const int C = (n + 1023) / 1024;
    const int begin = t * C;
    const int end   = (begin + C < n) ? (begin + C) : n;

    int s = 0;
    for (int i = begin; i < end; ++i) s += counts[i];
    partial[t] = s;
    __syncthreads();

    if (t == 0) {
        int acc = 0;
        for (int i = 0; i < 1024; ++i) { int v = partial[i]; partial[i] = acc; acc += v; }
        offsets[n] = acc;   // total == E
    }
    __syncthreads();

    int acc = partial[t];
    for (int i = begin; i < end; ++i) {
        offsets[i] = acc;
        cursors[i] = acc;
        acc += counts[i];
    }
}

__global__ void bucket_kernel(const int* __restrict__ edge_index,
                              int* __restrict__ cursors,
                              int* __restrict__ sorted_src, int E) {
    int i = blockIdx.x * blockDim.x + threadIdx.x;
    if (i < E) {
        int src = edge_index[i];
        int tgt = edge_index[E + i];
        int pos = atomicAdd(&cursors[tgt], 1);
        sorted_src[pos] = src;
    }
}

// Recover wave-relative LDS byte address from a generic (flat) shared pointer.
// ISA: LDS aperture mapping truncates to addr[31:0].
__device__ __forceinline__ uint32_t lds_addr_of(const void* p) {
    return (uint32_t)(uintptr_t)p;
}

// One wave32 per target node, async-pipelined gather:
//   global_load_async_to_lds_b128 stages the next source row (512 B across 32
//   lanes) into an LDS double buffer while the current row is summed from LDS.
//   ASYNCcnt ordering: loads complete in order, so s_wait_asynccnt 1 after
//   issuing row e+1 guarantees row e is resident.
__global__ void reduce_async_kernel(const float* __restrict__ emb,
                                    const int* __restrict__ offsets,
                                    const int* __restrict__ sorted_src,
                                    float* __restrict__ out, int n) {
    __shared__ float4 stage[WAVES_PER_BLOCK][2][32];   // 8 KB per block

    const int gid  = blockIdx.x * blockDim.x + threadIdx.x;
    const int node = gid >> 5;
    const int wib  = (threadIdx.x >> 5) & (WAVES_PER_BLOCK - 1);
    const int lane = threadIdx.x & 31;
    if (node >= n) return;                 // node uniform per wave: whole-wave exit

    const int beg = offsets[node];
    const int end = offsets[node + 1];
    const int deg = end - beg;

    float4 acc = make_float4(0.f, 0.f, 0.f, 0.f);

    if (deg > 0) {
        const uint32_t lds0 = lds_addr_of(&stage[wib][0][lane]);
        const uint32_t lds1 = lds_addr_of(&stage[wib][1][lane]);

        // Prologue: stage first row into buffer 0
        {
            int src = sorted_src[beg];
            uint64_t g = (uint64_t)(uintptr_t)(emb + (size_t)src * HIDDEN + lane * 4);
            asm volatile("global_load_async_to_lds_b128 %0, %1, off"
                         :: "v"(lds0), "v"(g) : "memory");
        }

        int buf = 0;
        for (int e = beg; e < end; ++e) {
            // keep the index stream warm (gfx1250 global_prefetch_b8)
            __builtin_prefetch(&sorted_src[e + 16], 0, 2);

            if (e + 1 < end) {
                int nsrc = sorted_src[e + 1];
                uint64_t g = (uint64_t)(uintptr_t)(emb + (size_t)nsrc * HIDDEN + lane * 4);
                uint32_t dst = buf ? lds0 : lds1;
                asm volatile("global_load_async_to_lds_b128 %0, %1, off"
                             :: "v"(dst), "v"(g) : "memory");
                asm volatile("s_wait_asynccnt 0x1" ::: "memory");  // row e resident
            } else {
                asm volatile("s_wait_asynccnt 0x0" ::: "memory");  // last row resident
            }

            float4 v = stage[wib][buf][lane];
            acc.x += v.x; acc.y += v.y; acc.z += v.z; acc.w += v.w;
            buf ^= 1;
        }
    }

    float inv = 1.0f / (float)(deg > 0 ? deg : 1);
    acc.x *= inv; acc.y *= inv; acc.z *= inv; acc.w *= inv;

    ((float4*)out)[(size_t)node * H4 + lane] = acc;
}

// ---------------------------------------------------------------------------
// Fallback path (if ws too small for CSR): direct atomic scatter-add.
// ws layout (floats): [0, N) float counts
// ---------------------------------------------------------------------------

__global__ void zero_out_kernel(float* __restrict__ out, float* __restrict__ cnt,
                                int total, int n) {
    int i = blockIdx.x * blockDim.x + threadIdx.x;
    if (i < total) out[i] = 0.f;
    if (i < n)     cnt[i] = 0.f;
}

__global__ void scatter_atomic_kernel(const float* __restrict__ emb,
                                      const int* __restrict__ edge_index,
                                      float* __restrict__ out,
                                      float* __restrict__ cnt, int E) {
    const int gid  = blockIdx.x * blockDim.x + threadIdx.x;
    const int edge = gid >> 5;
    const int lane = threadIdx.x & 31;
    if (edge >= E) return;

    int src = edge_index[edge];
    int tgt = edge_index[E + edge];

    const float4* embv = (const float4*)emb;
    float4 v = embv[(size_t)src * H4 + lane];
    float* dst = out + (size_t)tgt * HIDDEN + lane * 4;
    atomicAdd(dst + 0, v.x);
    atomicAdd(dst + 1, v.y);
    atomicAdd(dst + 2, v.z);
    atomicAdd(dst + 3, v.w);
    if (lane == 0) atomicAdd(&cnt[tgt], 1.0f);
}

__global__ void divide_kernel(float* __restrict__ out, const float* __restrict__ cnt, int n) {
    const int gid  = blockIdx.x * blockDim.x + threadIdx.x;   // one thread per float4
    const int node = gid >> 5;
    const int lane = threadIdx.x & 31;
    if (node >= n) return;
    float c = cnt[node];
    float inv = 1.0f / fmaxf(c, 1.0f);
    float4* ov = (float4*)out;
    float4 v = ov[(size_t)node * H4 + lane];
    v.x *= inv; v.y *= inv; v.z *= inv; v.w *= inv;
    ov[(size_t)node * H4 + lane] = v;
}

// ---------------------------------------------------------------------------

extern "C" void kernel_launch(void* const* d_in, const int* in_sizes, int n_in,
                              void* d_out, int out_size, void* d_ws, size_t ws_size,
                              hipStream_t stream) {
    // inputs: [0] node_ids (arange, unused), [1] edge_index [2,E] int32, [2] emb [N,128] f32
    const int*   edge_index = (const int*)d_in[1];
    const float* emb        = (const float*)d_in[2];
    float*       out        = (float*)d_out;

    const int E = in_sizes[1] / 2;
    const int N = N_NODES;

    const size_t csr_need = (size_t)(3 * N + 1 + E) * sizeof(int);

    if (ws_size >= csr_need) {
        int* ws_i      = (int*)d_ws;
        int* counts    = ws_i;
        int* offsets   = ws_i + N;
        int* cursors   = ws_i + 2 * N + 1;
        int* sortedsrc = ws_i + 3 * N + 1;

        zero_counts_kernel<<<(N + 255) / 256, 256, 0, stream>>>(counts, N);
        hist_kernel<<<(E + 255) / 256, 256, 0, stream>>>(edge_index, counts, E);
        scan_kernel<<<1, 1024, 0, stream>>>(counts, offsets, cursors, N);
        bucket_kernel<<<(E + 255) / 256, 256, 0, stream>>>(edge_index, cursors, sortedsrc, E);

        const long long waves_threads = (long long)N * 32;
        reduce_async_kernel<<<(int)((waves_threads + 255) / 256), 256, 0, stream>>>(
            emb, offsets, sortedsrc, out, N);
    } else {
        // Fallback: direct atomic scatter-add (needs only N floats of scratch)
        float* cnt = (float*)d_ws;
        const int total = N * HIDDEN;
        zero_out_kernel<<<(total + 255) / 256, 256, 0, stream>>>(out, cnt, total, N);
        const long long ethreads = (long long)E * 32;
        scatter_atomic_kernel<<<(int)((ethreads + 255) / 256), 256, 0, stream>>>(
            emb, edge_index, out, cnt, E);
        const long long nthreads = (long long)N * 32;
        divide_kernel<<<(int)((nthreads + 255) / 256), 256, 0, stream>>>(out, cnt, N);
    }
}